// TriSourceQueryRouter_52166672778162
// MI455X (gfx1250) — compile-verified
//
#include <hip/hip_runtime.h>
#include <hip/hip_bf16.h>
#include <math.h>

// ---------------------------------------------------------------------------
// TriSourceQueryRouter for MI455X (gfx1250, wave32, WMMA + async-to-LDS)
// ---------------------------------------------------------------------------
#define NL 20000
#define NP 1000
#define NG 256
#define NT (NL + NP + NG)      // 21256 rows per batch
#define BD 256                 // model dim D
#define NB 16                  // batch
#define KEEP 900
#define MT 3                   // row-tiles (of 16) per block
#define ROWS_PB (MT * 16)      // 48 rows per block
#define NBLK ((NT + ROWS_PB - 1) / ROWS_PB)   // 443 blocks per batch
#define ASTR 260               // Afeat LDS row stride (floats): 16B-aligned, bank-skewed

typedef __attribute__((ext_vector_type(2))) float v2f;
typedef __attribute__((ext_vector_type(8))) float v8f;

// float -> order-preserving uint (ascending)
__device__ __forceinline__ unsigned int f2sortable(float f) {
    unsigned int u = __float_as_uint(f);
    return (u & 0x80000000u) ? ~u : (u | 0x80000000u);
}

// decode global row -> (src, local idx)
__device__ __forceinline__ void decode_row(int g, int& src, int& li) {
    if (g < NL)            { src = 0; li = g; }
    else if (g < NL + NP)  { src = 1; li = g - NL; }
    else                   { src = 2; li = g - NL - NP; }
}

// transformed score + validity for one element
__device__ __forceinline__ void raw_score(int src, float raw, float& s, bool& valid) {
    if (src == 0)      { s = log1pf(raw); valid = raw > 0.0f; }
    else if (src == 1) { s = raw;         valid = raw > 0.0f; }
    else               { s = raw;         valid = true; }
}

// ---------------------------------------------------------------------------
// Kernel A: per (batch, source) masked min / range
// ---------------------------------------------------------------------------
__global__ void k_stats(const float* __restrict__ lidar_s,
                        const float* __restrict__ prop_s,
                        const float* __restrict__ glob_s,
                        float* __restrict__ stats) {
    const int src = blockIdx.x;   // 0..2
    const int b   = blockIdx.y;   // 0..15
    const int tid = threadIdx.x;  // 256 threads
    int n;
    const float* sp;
    if (src == 0)      { n = NL; sp = lidar_s + (size_t)b * NL; }
    else if (src == 1) { n = NP; sp = prop_s  + (size_t)b * NP; }
    else               { n = NG; sp = glob_s  + (size_t)b * NG; }

    float mn = INFINITY, mx = -INFINITY;
    for (int i = tid; i < n; i += 256) {
        float s; bool valid;
        raw_score(src, sp[i], s, valid);
        if (valid) { mn = fminf(mn, s); mx = fmaxf(mx, s); }
    }
    __shared__ float smn[256], smx[256];
    smn[tid] = mn; smx[tid] = mx;
    __syncthreads();
    for (int off = 128; off > 0; off >>= 1) {
        if (tid < off) {
            smn[tid] = fminf(smn[tid], smn[tid + off]);
            smx[tid] = fmaxf(smx[tid], smx[tid + off]);
        }
        __syncthreads();
    }
    if (tid == 0) {
        stats[(b * 3 + src) * 2 + 0] = smn[0];
        stats[(b * 3 + src) * 2 + 1] = smx[0] - smn[0];
    }
}

__device__ __forceinline__ float norm_score(const float* stats, int b, int src,
                                            float s, bool valid) {
    float mn  = stats[(b * 3 + src) * 2 + 0];
    float rng = stats[(b * 3 + src) * 2 + 1];
    float nrm = (fabsf(rng) < 1e-6f) ? 1.0f : (s - mn) / rng;
    return valid ? nrm : 0.0f;
}

// ---------------------------------------------------------------------------
// Kernel P: pack W1 rows 0..255 into WMMA-B fragment order.
// pack[kk][col][j] = W1[kk*4 + j][col]  -> lane (bf.x,bf.y) is one b64 load.
// ---------------------------------------------------------------------------
__global__ void k_packW1(const float* __restrict__ W1, float* __restrict__ pack) {
    const int kk  = blockIdx.x;   // 0..63
    const int col = threadIdx.x;  // 0..255
#pragma unroll
    for (int j = 0; j < 4; ++j)
        pack[((size_t)kk * 256 + col) * 4 + j] = W1[(size_t)(kk * 4 + j) * BD + col];
}

// ---------------------------------------------------------------------------
// Kernel E: fold source embeddings into the hidden bias:
// bias[src][col] = b1[col] + sum_k emb[src][k] * W1[k][col]
// (exact:  (q+emb)@W1 = q@W1 + emb@W1)
// ---------------------------------------------------------------------------
__global__ void k_bias(const float* __restrict__ emb, const float* __restrict__ W1,
                       const float* __restrict__ b1, float* __restrict__ bias) {
    const int src = blockIdx.x;   // 0..2
    const int col = threadIdx.x;  // 0..255
    float s = b1[col];
    for (int k = 0; k < BD; ++k)
        s += emb[src * BD + k] * W1[(size_t)k * BD + col];
    bias[src * BD + col] = s;
}

// ---------------------------------------------------------------------------
// Kernel B: score head via V_WMMA_F32_16X16X4_F32 with async-to-LDS staging.
// One block = 48 rows of one batch; 128 threads = 4 waves.
// ---------------------------------------------------------------------------
__global__ void __launch_bounds__(128)
k_score_head(const float* __restrict__ lidar_q,
             const float* __restrict__ prop_q,
             const float* __restrict__ glob_q,
             const float* __restrict__ lidar_s,
             const float* __restrict__ prop_s,
             const float* __restrict__ glob_s,
             const float* __restrict__ W1,     // [257,256] (row 256 used here)
             const float* __restrict__ packW1, // [64,256,4] fragment-packed
             const float* __restrict__ biasA,  // [3,256] emb-folded bias
             const float* __restrict__ W2,     // [256]
             const float* __restrict__ b2,     // [1]
             const float* __restrict__ stats,
             float* __restrict__ keep_out) {   // [NB, NT]
    const int blk = blockIdx.x;
    const int b   = blockIdx.y;
    const int tid = threadIdx.x;
    const int g0  = blk * ROWS_PB;

    __shared__ float Afeat[ROWS_PB][ASTR];   // raw query rows (16B-aligned rows)
    __shared__ float snormL[ROWS_PB];
    __shared__ float logitL[ROWS_PB];
    __shared__ const float* qrow[ROWS_PB];
    __shared__ int srow[ROWS_PB];

    // ---- row setup: pointers + normalized score (threads 0..47) ----
    if (tid < ROWS_PB) {
        const int g = g0 + tid;
        const float* qp;
        int src = 0;
        float nrm = 0.0f;
        if (g < NT) {
            int li; decode_row(g, src, li);
            float raw, s; bool valid;
            if (src == 0)      { raw = lidar_s[(size_t)b * NL + li]; qp = lidar_q + ((size_t)b * NL + li) * BD; }
            else if (src == 1) { raw = prop_s[(size_t)b * NP + li];  qp = prop_q  + ((size_t)b * NP + li) * BD; }
            else               { raw = glob_s[(size_t)b * NG + li];  qp = glob_q  + ((size_t)b * NG + li) * BD; }
            raw_score(src, raw, s, valid);
            nrm = norm_score(stats, b, src, s, valid);
        } else {
            qp = lidar_q + (size_t)b * NL * BD;   // dummy valid row; result unused
        }
        qrow[tid]   = qp;
        srow[tid]   = src;
        snormL[tid] = nrm;
        logitL[tid] = 0.0f;
    }
    __syncthreads();

    // ---- stage A tile via GLOBAL_LOAD_ASYNC_TO_LDS_B128 (ASYNCcnt) ----
    // 48 rows x 64 chunks of 16B; per-lane global address + LDS byte offset.
    {
        const unsigned int ldsBase = (unsigned int)(uintptr_t)(&Afeat[0][0]);
        for (int c = tid; c < ROWS_PB * 64; c += 128) {
            const int r  = c >> 6;
            const int q4 = c & 63;
            const unsigned long long ga =
                (unsigned long long)(uintptr_t)(qrow[r] + q4 * 4);
            const unsigned int la = ldsBase + r * (ASTR * 4) + q4 * 16;
            asm volatile("global_load_async_to_lds_b128 %0, %1, off"
                         :: "v"(la), "v"(ga) : "memory");
        }
        asm volatile("s_wait_asynccnt 0x0" ::: "memory");
    }
    __syncthreads();

    // ---- WMMA f32 16x16x4 main loop ----
    const int wave    = tid >> 5;
    const int lane    = tid & 31;
    const int laneRow = lane & 15;
    const int koff    = (lane >> 4) * 2;          // 0 or 2
    const int colBase = wave * 64 + (lane & 15);

    const float* bptr = packW1 + ((size_t)colBase * 4 + koff);

    v8f acc[MT][4];
#pragma unroll
    for (int m = 0; m < MT; ++m)
#pragma unroll
        for (int t = 0; t < 4; ++t) acc[m][t] = (v8f){};

    for (int k4 = 0; k4 < 64; ++k4) {
        const int kk = k4 * 4 + koff;
        v2f a[MT];
#pragma unroll
        for (int m = 0; m < MT; ++m) {
            a[m].x = Afeat[m * 16 + laneRow][kk];
            a[m].y = Afeat[m * 16 + laneRow][kk + 1];
        }
        const float* bk = bptr + (size_t)k4 * 1024;
#pragma unroll
        for (int t = 0; t < 4; ++t) {
            const v2f bf = *reinterpret_cast<const v2f*>(bk + t * 64);
#pragma unroll
            for (int m = 0; m < MT; ++m)
                acc[m][t] = __builtin_amdgcn_wmma_f32_16x16x4_f32(
                    false, a[m], false, bf, (short)0, acc[m][t], false, false);
        }
    }

    // ---- epilogue: +snorm*W1[256], +bias[src], relu, dot W2, reduce ----
    const int rowOff = (lane >= 16) ? 8 : 0;
#pragma unroll
    for (int m = 0; m < MT; ++m) {
        float sn[8]; int sv[8];
#pragma unroll
        for (int r = 0; r < 8; ++r) {
            const int row = m * 16 + r + rowOff;
            sn[r] = snormL[row];
            sv[r] = srow[row];
        }
        float psum[8];
#pragma unroll
        for (int r = 0; r < 8; ++r) psum[r] = 0.0f;
#pragma unroll
        for (int t = 0; t < 4; ++t) {
            const int colt = colBase + t * 16;
            const float w2v   = W2[colt];
            const float w1l   = W1[(size_t)256 * BD + colt];
            const float bias0 = biasA[0 * BD + colt];
            const float bias1 = biasA[1 * BD + colt];
            const float bias2 = biasA[2 * BD + colt];
#pragma unroll
            for (int r = 0; r < 8; ++r) {
                const float bv = (sv[r] == 0) ? bias0 : ((sv[r] == 1) ? bias1 : bias2);
                float h = acc[m][t][r] + sn[r] * w1l + bv;
                h = fmaxf(h, 0.0f);
                psum[r] += h * w2v;
            }
        }
        // reduce across the 16 lanes of each half-wave (masks 1,2,4,8 stay in-half)
#pragma unroll
        for (int msk = 1; msk < 16; msk <<= 1) {
#pragma unroll
            for (int r = 0; r < 8; ++r) psum[r] += __shfl_xor(psum[r], msk, 32);
        }
        if ((lane & 15) == 0) {
#pragma unroll
            for (int r = 0; r < 8; ++r) atomicAdd(&logitL[m * 16 + r + rowOff], psum[r]);
        }
    }
    __syncthreads();

    if (tid < ROWS_PB) {
        const int g = g0 + tid;
        if (g < NT) {
            keep_out[(size_t)b * NT + g] = logitL[tid] + b2[0] + snormL[tid];
        }
    }
}

// ---------------------------------------------------------------------------
// Kernel C: top-900 per batch, threshold bin taken first-come (sorted desc, ties by ascending index)
// ---------------------------------------------------------------------------
__global__ void __launch_bounds__(256)
k_select(const float* __restrict__ lidar_s,
         const float* __restrict__ prop_s,
         const float* __restrict__ keep,      // [NB, NT]
         unsigned int* __restrict__ topk) {   // [NB, KEEP]
    const int b   = blockIdx.x;
    const int tid = threadIdx.x;

    __shared__ unsigned int hist[4096];
    __shared__ unsigned long long sel[1024];
    __shared__ int Tbin, needT, cntSel, takenT;

    for (int i = tid; i < 4096; i += 256) hist[i] = 0u;
    if (tid == 0) { cntSel = 0; takenT = 0; }
    __syncthreads();

    // pass 1: histogram of valid keys
    for (int n = tid; n < NT; n += 256) {
        bool valid;
        if (n < NL)            valid = lidar_s[(size_t)b * NL + n] > 0.0f;
        else if (n < NL + NP)  valid = prop_s[(size_t)b * NP + (n - NL)] > 0.0f;
        else                   valid = true;
        if (valid) {
            unsigned int u = f2sortable(keep[(size_t)b * NT + n]);
            atomicAdd(&hist[u >> 20], 1u);
        }
    }
    __syncthreads();

    if (tid == 0) {
        int cum = 0, tb = -1, nd = 0;
        for (int bi = 4095; bi >= 0; --bi) {
            int h = (int)hist[bi];
            if (cum + h >= KEEP) { tb = bi; nd = KEEP - cum; break; }
            cum += h;
        }
        Tbin = tb; needT = nd;
    }
    __syncthreads();

    // pass 2: collect candidates
    const int tb = Tbin, nd = needT;
    for (int n = tid; n < NT; n += 256) {
        bool valid;
        if (n < NL)            valid = lidar_s[(size_t)b * NL + n] > 0.0f;
        else if (n < NL + NP)  valid = prop_s[(size_t)b * NP + (n - NL)] > 0.0f;
        else                   valid = true;
        if (!valid) continue;
        unsigned int u = f2sortable(keep[(size_t)b * NT + n]);
        int bin = (int)(u >> 20);
        bool take = false;
        if (bin > tb)       take = true;
        else if (bin == tb) take = (atomicAdd(&takenT, 1) < nd);
        if (take) {
            int s = atomicAdd(&cntSel, 1);
            if (s < 1024) {
                sel[s] = ((unsigned long long)u << 32) |
                         (unsigned long long)(0xFFFFFFFFu - (unsigned int)n);
            }
        }
    }
    __syncthreads();
    // pad with sentinel (smallest key, decodes to index 0)
    for (int s = cntSel + tid; s < 1024; s += 256) sel[s] = 0xFFFFFFFFull;
    __syncthreads();

    // bitonic sort, descending
    for (int k = 2; k <= 1024; k <<= 1) {
        for (int j = k >> 1; j > 0; j >>= 1) {
            for (int i = tid; i < 1024; i += 256) {
                int ixj = i ^ j;
                if (ixj > i) {
                    unsigned long long A = sel[i], Bv = sel[ixj];
                    bool desc = ((i & k) == 0);
                    bool sw = desc ? (A < Bv) : (A > Bv);
                    if (sw) { sel[i] = Bv; sel[ixj] = A; }
                }
            }
            __syncthreads();
        }
    }

    for (int s = tid; s < KEEP; s += 256) {
        unsigned int n = 0xFFFFFFFFu - (unsigned int)(sel[s] & 0xFFFFFFFFull);
        topk[(size_t)b * KEEP + s] = n;
    }
}

// ---------------------------------------------------------------------------
// Kernel D: gather routed outputs into concatenated d_out (all as f32 values)
// ---------------------------------------------------------------------------
__global__ void __launch_bounds__(256)
k_gather(const float* __restrict__ lidar_q, const float* __restrict__ lidar_r,
         const float* __restrict__ lidar_s, const int* __restrict__ lidar_lab,
         const float* __restrict__ lidar_ps, const unsigned char* __restrict__ lidar_pv,
         const float* __restrict__ prop_q, const float* __restrict__ prop_r,
         const float* __restrict__ prop_s,
         const float* __restrict__ glob_q, const float* __restrict__ glob_r,
         const float* __restrict__ emb,
         const float* __restrict__ keep,
         const unsigned int* __restrict__ topk,
         float* __restrict__ out) {
    const int s = blockIdx.x;     // 0..KEEP-1
    const int b = blockIdx.y;     // 0..NB-1
    const int t = threadIdx.x;    // 256

    const unsigned int g = topk[(size_t)b * KEEP + s];
    int src, li; decode_row((int)g, src, li);

    const float* qp; const float* rp;
    if (src == 0)      { qp = lidar_q + ((size_t)b * NL + li) * BD; rp = lidar_r + ((size_t)b * NL + li) * 3; }
    else if (src == 1) { qp = prop_q  + ((size_t)b * NP + li) * BD; rp = prop_r  + ((size_t)b * NP + li) * 3; }
    else               { qp = glob_q  + ((size_t)b * NG + li) * BD; rp = glob_r  + ((size_t)b * NG + li) * 3; }

    const size_t row = (size_t)b * KEEP + s;
    // routed_queries (= query + source embedding)
    out[row * BD + t] = qp[t] + emb[src * BD + t];

    const size_t o_refs   = (size_t)NB * KEEP * BD;
    const size_t o_scores = o_refs   + (size_t)NB * KEEP * 3;
    const size_t o_ids    = o_scores + (size_t)NB * KEEP;
    const size_t o_lab    = o_ids    + (size_t)NB * KEEP;
    const size_t o_ps     = o_lab    + (size_t)NB * KEEP;
    const size_t o_pv     = o_ps     + (size_t)NB * KEEP;
    const size_t o_val    = o_pv     + (size_t)NB * KEEP;

    if (t < 3) out[o_refs + row * 3 + t] = rp[t];
    if (t == 0) {
        out[o_scores + row] = keep[(size_t)b * NT + g];
        out[o_ids    + row] = (float)src;
        out[o_lab    + row] = (src == 0) ? (float)lidar_lab[(size_t)b * NL + li] : 0.0f;
        out[o_ps     + row] = (src == 0) ? lidar_ps[(size_t)b * NL + li] : 0.0f;
        out[o_pv     + row] = (src == 0) ? (lidar_pv[(size_t)b * NL + li] ? 1.0f : 0.0f) : 0.0f;
        bool valid;
        if (src == 0)      valid = lidar_s[(size_t)b * NL + li] > 0.0f;
        else if (src == 1) valid = prop_s[(size_t)b * NP + li] > 0.0f;
        else               valid = true;
        out[o_val + row] = valid ? 1.0f : 0.0f;
    }
}

// ---------------------------------------------------------------------------
extern "C" void kernel_launch(void* const* d_in, const int* in_sizes, int n_in,
                              void* d_out, int out_size, void* d_ws, size_t ws_size,
                              hipStream_t stream) {
    const float* lidar_q   = (const float*)d_in[0];
    const float* lidar_r   = (const float*)d_in[1];
    const float* lidar_s   = (const float*)d_in[2];
    const int*   lidar_lab = (const int*)d_in[3];
    const float* lidar_ps  = (const float*)d_in[4];
    const unsigned char* lidar_pv = (const unsigned char*)d_in[5];
    const float* prop_q    = (const float*)d_in[6];
    const float* prop_r    = (const float*)d_in[7];
    const float* prop_s    = (const float*)d_in[8];
    const float* glob_q    = (const float*)d_in[9];
    const float* glob_r    = (const float*)d_in[10];
    const float* glob_s    = (const float*)d_in[11];
    const float* emb       = (const float*)d_in[12];
    const float* W1        = (const float*)d_in[13];
    const float* b1        = (const float*)d_in[14];
    const float* W2        = (const float*)d_in[15];
    const float* b2        = (const float*)d_in[16];

    float* ws_f  = (float*)d_ws;
    float* stats = ws_f;                               // 96 floats
    float* keep  = ws_f + 96;                          // NB*NT floats
    unsigned int* topk = (unsigned int*)(keep + (size_t)NB * NT); // NB*KEEP
    float* packW1 = (float*)(topk + (size_t)NB * KEEP);           // 64*256*4 floats
    float* biasA  = packW1 + (size_t)64 * 256 * 4;                // 3*256 floats

    k_stats<<<dim3(3, NB), 256, 0, stream>>>(lidar_s, prop_s, glob_s, stats);

    k_packW1<<<dim3(64), 256, 0, stream>>>(W1, packW1);

    k_bias<<<dim3(3), 256, 0, stream>>>(emb, W1, b1, biasA);

    k_score_head<<<dim3(NBLK, NB), 128, 0, stream>>>(
        lidar_q, prop_q, glob_q, lidar_s, prop_s, glob_s,
        W1, packW1, biasA, W2, b2, stats, keep);

    k_select<<<dim3(NB), 256, 0, stream>>>(lidar_s, prop_s, keep, topk);

    k_gather<<<dim3(KEEP, NB), 256, 0, stream>>>(
        lidar_q, lidar_r, lidar_s, lidar_lab, lidar_ps, lidar_pv,
        prop_q, prop_r, prop_s, glob_q, glob_r,
        emb, keep, topk, (float*)d_out);
}